// LinearAttention_25091198943928
// MI455X (gfx1250) — compile-verified
//
#include <hip/hip_runtime.h>

typedef float v2f __attribute__((ext_vector_type(2)));
typedef float v8f __attribute__((ext_vector_type(8)));

#define NB 2
#define NC 64
#define NHID 128
#define NN (48*48*48)   // 110592

// workspace layout (floats)
#define WS_M  0                     // 256   row maxima
#define WS_S  256                   // 256   row exp-sums
#define WS_G  512                   // 16384 G = P x^T
#define WS_B  16896                 // 8192  folded 64x64 per batch
#define WS_PS 25088                 // 2*108*128 partial sums
#define WS_PG 52736                 // 2*108*8192 partial G
#define K2_WGS 108

__device__ __forceinline__ v8f vzero() {
  v8f z = {0.f,0.f,0.f,0.f,0.f,0.f,0.f,0.f};
  return z;
}

__device__ __forceinline__ v8f wmma4(v2f a, v2f b, v8f c) {
  // D = A(16x4 f32) * B(4x16 f32) + C(16x16 f32)
  return __builtin_amdgcn_wmma_f32_16x16x4_f32(false, a, false, b, (short)0, c, false, false);
}

__device__ __forceinline__ void atomicMaxF(float* addr, float val) {
  int old = __float_as_int(*addr);
  while (__int_as_float(old) < val) {
    int prev = atomicCAS((int*)addr, old, __float_as_int(val));
    if (prev == old) break;
    old = prev;
  }
}

__global__ __launch_bounds__(256) void k_init(float* __restrict__ m) {
  const int i = blockIdx.x * 256 + threadIdx.x;
  if (i < NB * NHID) m[i] = -3.0e38f;
}

// ---------------- Pass 1: row-max of k = W_k x --------------------------
__global__ __launch_bounds__(256) void k_rowmax(
    const float* __restrict__ x, const float* __restrict__ wqkv,
    float* __restrict__ m) {
  const int b = blockIdx.y;
  const float* xb = x + (size_t)b * NC * NN;
  const int wv  = threadIdx.x >> 5;
  const int lane = threadIdx.x & 31;
  const int hl  = lane >> 4;
  const int l16 = lane & 15;

  v8f kmax[8];
#pragma unroll
  for (int t = 0; t < 8; ++t) kmax[t] = vzero();
#pragma unroll
  for (int t = 0; t < 8; ++t)
#pragma unroll
    for (int j = 0; j < 8; ++j) kmax[t][j] = -3.0e38f;

  const int stride = gridDim.x * 128;
  for (int n0 = blockIdx.x * 128 + wv * 16; n0 < NN; n0 += stride) {
    v2f Bf[16];
#pragma unroll
    for (int ks = 0; ks < 16; ++ks) {
      const int k = 4 * ks + 2 * hl;
      Bf[ks].x = xb[(size_t)k * NN + n0 + l16];
      Bf[ks].y = xb[(size_t)(k + 1) * NN + n0 + l16];
    }
#pragma unroll
    for (int t = 0; t < 8; ++t) {
      v8f acc = vzero();
#pragma unroll
      for (int ks = 0; ks < 16; ++ks) {
        const int r = 128 + 16 * t + l16;   // k rows of w_qkv
        v2f Af;
        Af.x = wqkv[r * 64 + 4 * ks + 2 * hl];
        Af.y = wqkv[r * 64 + 4 * ks + 2 * hl + 1];
        acc = wmma4(Af, Bf[ks], acc);
      }
#pragma unroll
      for (int j = 0; j < 8; ++j) kmax[t][j] = fmaxf(kmax[t][j], acc[j]);
    }
  }
#pragma unroll
  for (int t = 0; t < 8; ++t)
#pragma unroll
    for (int j = 0; j < 8; ++j) {
      float v = kmax[t][j];
      for (int off = 1; off < 16; off <<= 1)
        v = fmaxf(v, __shfl_xor(v, off, 32));
      if (l16 == 0) atomicMaxF(&m[b * NHID + 16 * t + j + 8 * hl], v);
    }
}

// ---- Pass 2: per-WG partial s = sum(exp(k-m)), G = exp(k-m) * x^T ------
__global__ __launch_bounds__(256) void k_stats(
    const float* __restrict__ x, const float* __restrict__ wqkv,
    const float* __restrict__ m, float* __restrict__ ps_part,
    float* __restrict__ pg_part) {
  __shared__ float xs[64 * 65];    // x tile, padded stride 65
  __shared__ float ps[128 * 65];   // P tile, padded stride 65
  const int b = blockIdx.y;
  const float* xb = x + (size_t)b * NC * NN;
  const int wv  = threadIdx.x >> 5;
  const int lane = threadIdx.x & 31;
  const int hl  = lane >> 4;
  const int l16 = lane & 15;

  v2f Af[16];                        // W_k frags for this wave's 16 rows
#pragma unroll
  for (int ks = 0; ks < 16; ++ks) {
    const int r = 128 + 16 * wv + l16;
    Af[ks].x = wqkv[r * 64 + 4 * ks + 2 * hl];
    Af[ks].y = wqkv[r * 64 + 4 * ks + 2 * hl + 1];
  }
  float mrow[8];
#pragma unroll
  for (int j = 0; j < 8; ++j) mrow[j] = m[b * NHID + 16 * wv + j + 8 * hl];

  v8f ssum = vzero();
  v8f Gacc[4];
#pragma unroll
  for (int ct = 0; ct < 4; ++ct) Gacc[ct] = vzero();

  const int stride = gridDim.x * 64;
  for (int n0 = blockIdx.x * 64; n0 < NN; n0 += stride) {
    for (int idx = threadIdx.x; idx < 64 * 64; idx += 256) {
      const int c = idx >> 6, col = idx & 63;
      xs[c * 65 + col] = xb[(size_t)c * NN + n0 + col];
    }
    __syncthreads();
    // k-GEMM for rows [16wv,16wv+16) x 64 cols, then exp -> ps
#pragma unroll
    for (int st = 0; st < 4; ++st) {
      v8f acc = vzero();
#pragma unroll
      for (int ks = 0; ks < 16; ++ks) {
        const int k = 4 * ks + 2 * hl;
        v2f Bf;
        Bf.x = xs[k * 65 + st * 16 + l16];
        Bf.y = xs[(k + 1) * 65 + st * 16 + l16];
        acc = wmma4(Af[ks], Bf, acc);
      }
#pragma unroll
      for (int j = 0; j < 8; ++j) {
        const float p = __expf(acc[j] - mrow[j]);
        ssum[j] += p;
        ps[(16 * wv + j + 8 * hl) * 65 + st * 16 + l16] = p;
      }
    }
    __syncthreads();
    // G-GEMM: Gacc += P(16x64) * X^T(64x64), K axis = n
#pragma unroll
    for (int ks = 0; ks < 16; ++ks) {
      const int k = 4 * ks + 2 * hl;
      v2f Ap;
      Ap.x = ps[(16 * wv + l16) * 65 + k];
      Ap.y = ps[(16 * wv + l16) * 65 + k + 1];
#pragma unroll
      for (int ct = 0; ct < 4; ++ct) {
        v2f Bx;
        Bx.x = xs[(ct * 16 + l16) * 65 + k];
        Bx.y = xs[(ct * 16 + l16) * 65 + k + 1];
        Gacc[ct] = wmma4(Ap, Bx, Gacc[ct]);
      }
    }
    __syncthreads();
  }
#pragma unroll
  for (int j = 0; j < 8; ++j) {
    float v = ssum[j];
    for (int off = 1; off < 16; off <<= 1) v += __shfl_xor(v, off, 32);
    if (l16 == 0)
      ps_part[(b * K2_WGS + blockIdx.x) * NHID + 16 * wv + j + 8 * hl] = v;
  }
#pragma unroll
  for (int ct = 0; ct < 4; ++ct)
#pragma unroll
    for (int j = 0; j < 8; ++j)
      pg_part[((size_t)(b * K2_WGS + blockIdx.x) * NHID + 16 * wv + j + 8 * hl) * 64
              + ct * 16 + l16] = Gacc[ct][j];
}

// ---------------- deterministic fixed-order reduction -------------------
__global__ __launch_bounds__(256) void k_reduce(
    const float* __restrict__ ps_part, const float* __restrict__ pg_part,
    float* __restrict__ s, float* __restrict__ G) {
  const int i = blockIdx.x * 256 + threadIdx.x;
  const int NS = NB * NHID;
  const int NG = NB * NHID * 64;
  if (i < NS) {
    const int b = i / NHID, r = i % NHID;
    float acc = 0.f;
    for (int w = 0; w < K2_WGS; ++w)
      acc += ps_part[(b * K2_WGS + w) * NHID + r];
    s[i] = acc;
  } else if (i < NS + NG) {
    const int j = i - NS;
    const int b = j / (NHID * 64), rc = j % (NHID * 64);
    float acc = 0.f;
    for (int w = 0; w < K2_WGS; ++w)
      acc += pg_part[(size_t)(b * K2_WGS + w) * NHID * 64 + rc];
    G[j] = acc;
  }
}

// ------- fold context, W_q, W_out into one 64x64 matrix per batch -------
__global__ __launch_bounds__(256) void k_combine(
    const float* __restrict__ G, const float* __restrict__ s,
    const float* __restrict__ wqkv, const float* __restrict__ wout,
    float* __restrict__ Bm) {
  __shared__ float ctx[4096];      // [h][d][e]
  __shared__ float Am[8192];       // [hid][c]
  const int b = blockIdx.x;
  const float* Gb = G + b * NHID * 64;
  const float* sb = s + b * NHID;
  for (int idx = threadIdx.x; idx < 4096; idx += 256) {
    const int hd = idx >> 5, e = idx & 31, h = hd >> 5;
    const float inv = 1.0f / sb[hd];
    const float* grow = Gb + hd * 64;
    const float* vrow = wqkv + (256 + h * 32 + e) * 64;   // W_v rows
    float acc = 0.f;
    for (int c = 0; c < 64; ++c) acc += grow[c] * vrow[c];
    ctx[idx] = acc * inv;
  }
  __syncthreads();
  for (int idx = threadIdx.x; idx < 8192; idx += 256) {
    const int hid = idx >> 6, c = idx & 63;
    const int h = hid >> 5, e = hid & 31;
    float acc = 0.f;
    for (int d = 0; d < 32; ++d)
      acc += ctx[(h * 32 + d) * 32 + e] * wqkv[(h * 32 + d) * 64 + c]; // W_q rows
    Am[idx] = acc;
  }
  __syncthreads();
  for (int idx = threadIdx.x; idx < 4096; idx += 256) {
    const int o = idx >> 6, c = idx & 63;
    float acc = 0.f;
    for (int hid = 0; hid < 128; ++hid)
      acc += wout[o * 128 + hid] * Am[hid * 64 + c];
    Bm[b * 4096 + idx] = acc;
  }
}

// ---------------- Pass 3: out = Bm @ x + bias ---------------------------
__global__ __launch_bounds__(256) void k_out(
    const float* __restrict__ x, const float* __restrict__ Bm,
    const float* __restrict__ bias, float* __restrict__ out) {
  const int b = blockIdx.y;
  const float* xb = x + (size_t)b * NC * NN;
  float* ob = out + (size_t)b * NC * NN;
  const float* Bb = Bm + b * 4096;
  const int wv  = threadIdx.x >> 5;
  const int lane = threadIdx.x & 31;
  const int hl  = lane >> 4;
  const int l16 = lane & 15;

  float binit[4][8];
#pragma unroll
  for (int t = 0; t < 4; ++t)
#pragma unroll
    for (int j = 0; j < 8; ++j) binit[t][j] = bias[16 * t + j + 8 * hl];

  const int stride = gridDim.x * 128;
  for (int n0 = blockIdx.x * 128 + wv * 16; n0 < NN; n0 += stride) {
    v2f Bf[16];
#pragma unroll
    for (int ks = 0; ks < 16; ++ks) {
      const int k = 4 * ks + 2 * hl;
      Bf[ks].x = xb[(size_t)k * NN + n0 + l16];
      Bf[ks].y = xb[(size_t)(k + 1) * NN + n0 + l16];
    }
#pragma unroll
    for (int t = 0; t < 4; ++t) {
      v8f acc = vzero();
#pragma unroll
      for (int j = 0; j < 8; ++j) acc[j] = binit[t][j];
#pragma unroll
      for (int ks = 0; ks < 16; ++ks) {
        const int r = 16 * t + l16;
        v2f Af;
        Af.x = Bb[r * 64 + 4 * ks + 2 * hl];
        Af.y = Bb[r * 64 + 4 * ks + 2 * hl + 1];
        acc = wmma4(Af, Bf[ks], acc);
      }
#pragma unroll
      for (int j = 0; j < 8; ++j)
        ob[(size_t)(16 * t + j + 8 * hl) * NN + n0 + l16] = acc[j];
    }
  }
}

extern "C" void kernel_launch(void* const* d_in, const int* in_sizes, int n_in,
                              void* d_out, int out_size, void* d_ws, size_t ws_size,
                              hipStream_t stream) {
  const float* x    = (const float*)d_in[0];
  const float* wqkv = (const float*)d_in[1];
  const float* wout = (const float*)d_in[2];
  const float* bout = (const float*)d_in[3];
  float* out = (float*)d_out;
  float* ws  = (float*)d_ws;

  k_init<<<1, 256, 0, stream>>>(ws + WS_M);
  k_rowmax<<<dim3(216, NB), 256, 0, stream>>>(x, wqkv, ws + WS_M);
  k_stats<<<dim3(K2_WGS, NB), 256, 0, stream>>>(x, wqkv, ws + WS_M,
                                                ws + WS_PS, ws + WS_PG);
  k_reduce<<<65, 256, 0, stream>>>(ws + WS_PS, ws + WS_PG, ws + WS_S, ws + WS_G);
  k_combine<<<NB, 256, 0, stream>>>(ws + WS_G, ws + WS_S, wqkv, wout, ws + WS_B);
  k_out<<<dim3(216, NB), 256, 0, stream>>>(x, ws + WS_B, bout, out);
}